// GATv2_84413287235623
// MI455X (gfx1250) — compile-verified
//
#include <hip/hip_runtime.h>
#include <hip/hip_bf16.h>
#include <cstdint>

#define N_NODES  10000
#define N_EDGES  160000
#define N_GRAPHS 64
#define DIM_IN   771
#define HIDC     128
#define NEG_SLOPE 0.2f
#define LN_EPS    1e-5f

typedef __attribute__((ext_vector_type(2))) float v2f;
typedef __attribute__((ext_vector_type(8))) float v8f;

// ------------------------------------------------------------------
// fp32 WMMA GEMM: out[nrows,M] = X[nrows,K] @ W[K,M] + bias (opt ReLU)
// One wave computes one 16x16 tile via V_WMMA_F32_16X16X4_F32.
// Main K-loop is guard-free (no exec masking around loads); the K%4
// tail uses clamped addresses + selects so EXEC stays all-ones.
// nrows must be a multiple of 16 (10000 and 64 both are).
// ------------------------------------------------------------------
__global__ void wmma_gemm_bias(const float* __restrict__ X, const float* __restrict__ W,
                               const float* __restrict__ bias, float* __restrict__ out,
                               int nrows, int K, int M, int relu) {
  const int lane = threadIdx.x & 31;
  const int wave = threadIdx.x >> 5;
  const int half = lane >> 4;   // 0: K pair (k,k+1), 1: (k+2,k+3)
  const int l16  = lane & 15;
  const int col0 = (blockIdx.x * 4 + wave) * 16;
  const int row0 = blockIdx.y * 16;
  if (col0 >= M || row0 >= nrows) return;   // wave-uniform
  const float* __restrict__ Xr = X + (size_t)(row0 + l16) * K;   // A row for this lane
  const float* __restrict__ Wc = W + (col0 + l16);               // B column for this lane
  const int K4 = K & ~3;
  v8f acc = {};
  int k = 0;
#pragma unroll 8
  for (; k < K4; k += 4) {
    const int ka = k + half * 2;
    v2f a, b;
    a.x = Xr[ka];
    a.y = Xr[ka + 1];
    b.x = Wc[(size_t)ka * M];
    b.y = Wc[(size_t)(ka + 1) * M];
    acc = __builtin_amdgcn_wmma_f32_16x16x4_f32(false, a, false, b,
                                                (short)0, acc, false, false);
  }
  if (K4 < K) {                 // wave-uniform scalar branch
    const int ka = k + half * 2;
    const int i0 = (ka     < K) ? ka       : (K - 1);  // clamped, always-valid
    const int i1 = (ka + 1 < K) ? (ka + 1) : (K - 1);
    const float ax = Xr[i0], ay = Xr[i1];
    const float bx = Wc[(size_t)i0 * M], by = Wc[(size_t)i1 * M];
    v2f a, b;
    a.x = (ka     < K) ? ax : 0.f;   // v_cndmask, no exec masking
    a.y = (ka + 1 < K) ? ay : 0.f;
    b.x = (ka     < K) ? bx : 0.f;
    b.y = (ka + 1 < K) ? by : 0.f;
    acc = __builtin_amdgcn_wmma_f32_16x16x4_f32(false, a, false, b,
                                                (short)0, acc, false, false);
  }
  const float bv = bias[col0 + l16];
#pragma unroll
  for (int i = 0; i < 8; ++i) {
    const int r = row0 + i + half * 8;
    float v = acc[i] + bv;
    if (relu) v = fmaxf(v, 0.f);
    out[(size_t)r * M + (col0 + l16)] = v;
  }
}

// ------------------------------------------------------------------
// float atomic max via int atomics (sign-aware, safe for mixed signs)
// ------------------------------------------------------------------
__device__ inline void atomicMaxF(float* addr, float value) {
  if (value >= 0.f) atomicMax((int*)addr, __float_as_int(value));
  else              atomicMin((unsigned int*)addr, __float_as_uint(value));
}

__device__ inline float leaky(float v) { return (v > 0.f) ? v : NEG_SLOPE * v; }

__global__ void fill_f32(float* __restrict__ p, float v, size_t n) {
  size_t i = (size_t)blockIdx.x * blockDim.x + threadIdx.x;
  if (i < n) p[i] = v;
}

// ---- edge pass A: logits + segment max (one wave per edge; float4 lanes) ----
__global__ void edge_logits_kernel(const int* __restrict__ ei,
                                   const float* __restrict__ xl,
                                   const float* __restrict__ xr,
                                   const float* __restrict__ att,
                                   float* __restrict__ logit,
                                   float* __restrict__ mmax,
                                   int E_, int NN, int H, int C) {
  const int e    = blockIdx.x * (blockDim.x >> 5) + (threadIdx.x >> 5);
  const int lane = threadIdx.x & 31;
  const int EP   = E_ + NN;
  if (e >= EP) return;
  const int sn = (e < E_) ? ei[e]      : (e - E_);
  const int dn = (e < E_) ? ei[E_ + e] : (e - E_);
  const float4* __restrict__ pl4 = (const float4*)(xl + (size_t)sn * H * C);
  const float4* __restrict__ pr4 = (const float4*)(xr + (size_t)dn * H * C);
  const float4* __restrict__ at4 = (const float4*)att;
  const int Cq = C >> 2;   // C==128 -> 32: one float4 per lane per head
  for (int h = 0; h < H; ++h) {
    float s = 0.f;
    for (int c4 = lane; c4 < Cq; c4 += 32) {
      const float4 l = pl4[h * Cq + c4];
      const float4 r = pr4[h * Cq + c4];
      const float4 a = at4[h * Cq + c4];
      s += leaky(l.x + r.x) * a.x + leaky(l.y + r.y) * a.y +
           leaky(l.z + r.z) * a.z + leaky(l.w + r.w) * a.w;
    }
#pragma unroll
    for (int off = 16; off; off >>= 1) s += __shfl_xor(s, off, 32);
    if (lane == 0) {
      logit[(size_t)e * H + h] = s;
      atomicMaxF(&mmax[(size_t)dn * H + h], s);
    }
  }
}

// ---- edge pass B: p = exp(logit - max); denom += p ----
__global__ void edge_softmax_kernel(const int* __restrict__ ei,
                                    const float* __restrict__ mmax,
                                    float* __restrict__ pbuf,
                                    float* __restrict__ den,
                                    int E_, int NN, int H) {
  const int idx = blockIdx.x * blockDim.x + threadIdx.x;
  const int EP  = E_ + NN;
  if (idx >= EP * H) return;
  const int e = idx / H, h = idx - e * H;
  const int dn = (e < E_) ? ei[E_ + e] : (e - E_);
  const float p = __expf(pbuf[idx] - mmax[(size_t)dn * H + h]);
  pbuf[idx] = p;
  atomicAdd(&den[(size_t)dn * H + h], p);
}

// ---- edge pass C: agg[dst] += alpha * xl[src] (one wave per edge; float4) ----
__global__ void edge_aggregate_kernel(const int* __restrict__ ei,
                                      const float* __restrict__ xl,
                                      const float* __restrict__ pbuf,
                                      const float* __restrict__ den,
                                      float* __restrict__ agg,
                                      int E_, int NN, int H, int C) {
  const int e    = blockIdx.x * (blockDim.x >> 5) + (threadIdx.x >> 5);
  const int lane = threadIdx.x & 31;
  const int EP   = E_ + NN;
  if (e >= EP) return;
  const int sn = (e < E_) ? ei[e]      : (e - E_);
  const int dn = (e < E_) ? ei[E_ + e] : (e - E_);
  float alpha[4];
  for (int h = 0; h < H; ++h)
    alpha[h] = pbuf[(size_t)e * H + h] / den[(size_t)dn * H + h];
  const float4* __restrict__ pl4 = (const float4*)(xl + (size_t)sn * H * C);
  float*        __restrict__ po  = agg + (size_t)dn * H * C;
  const int Dq = (H * C) >> 2;
  for (int j4 = lane; j4 < Dq; j4 += 32) {
    const float4 v = pl4[j4];
    const float  al = alpha[(j4 << 2) / C];   // 4 elems always within one head
    float* d = po + (j4 << 2);
    atomicAdd(d + 0, al * v.x);
    atomicAdd(d + 1, al * v.y);
    atomicAdd(d + 2, al * v.z);
    atomicAdd(d + 3, al * v.w);
  }
}

// ---- bias + LayerNorm + ELU, one wave per node (float4 lanes) ----
__global__ void node_ln_elu_kernel(const float* __restrict__ agg,
                                   const float* __restrict__ bo,
                                   const float* __restrict__ g,
                                   const float* __restrict__ be,
                                   float* __restrict__ xout, int n, int D) {
  const int node = blockIdx.x * (blockDim.x >> 5) + (threadIdx.x >> 5);
  const int lane = threadIdx.x & 31;
  if (node >= n) return;
  const float4* __restrict__ pin = (const float4*)(agg + (size_t)node * D);
  const float4* __restrict__ bo4 = (const float4*)bo;
  const float4* __restrict__ g4  = (const float4*)g;
  const float4* __restrict__ be4 = (const float4*)be;
  float4*       __restrict__ po  = (float4*)(xout + (size_t)node * D);
  const int Dq = D >> 2;
  float s = 0.f;
  for (int j = lane; j < Dq; j += 32) {
    const float4 v = pin[j]; const float4 b = bo4[j];
    s += (v.x + b.x) + (v.y + b.y) + (v.z + b.z) + (v.w + b.w);
  }
#pragma unroll
  for (int off = 16; off; off >>= 1) s += __shfl_xor(s, off, 32);
  const float mu = s / (float)D;
  float var = 0.f;
  for (int j = lane; j < Dq; j += 32) {
    const float4 v = pin[j]; const float4 b = bo4[j];
    float d0 = v.x + b.x - mu, d1 = v.y + b.y - mu, d2 = v.z + b.z - mu, d3 = v.w + b.w - mu;
    var += d0 * d0 + d1 * d1 + d2 * d2 + d3 * d3;
  }
#pragma unroll
  for (int off = 16; off; off >>= 1) var += __shfl_xor(var, off, 32);
  const float rstd = rsqrtf(var / (float)D + LN_EPS);
  for (int j = lane; j < Dq; j += 32) {
    const float4 v = pin[j]; const float4 b = bo4[j];
    const float4 gg = g4[j]; const float4 bb = be4[j];
    float4 o;
    float h0 = (v.x + b.x - mu) * rstd * gg.x + bb.x;
    float h1 = (v.y + b.y - mu) * rstd * gg.y + bb.y;
    float h2 = (v.z + b.z - mu) * rstd * gg.z + bb.z;
    float h3 = (v.w + b.w - mu) * rstd * gg.w + bb.w;
    o.x = (h0 > 0.f) ? h0 : (__expf(h0) - 1.f);
    o.y = (h1 > 0.f) ? h1 : (__expf(h1) - 1.f);
    o.z = (h2 > 0.f) ? h2 : (__expf(h2) - 1.f);
    o.w = (h3 > 0.f) ? h3 : (__expf(h3) - 1.f);
    po[j] = o;
  }
}

// ---- global mean pool (float4 per thread) ----
__global__ void pool_sum_kernel(const float* __restrict__ x, const int* __restrict__ batch,
                                float* __restrict__ sums, float* __restrict__ cnt,
                                int n, int D) {
  const int Dq = D >> 2;
  const int i = blockIdx.x * blockDim.x + threadIdx.x;   // over n * Dq
  if (i >= n * Dq) return;
  const int node = i / Dq, c4 = i - node * Dq;
  const int gidx = batch[node];
  const float4 v = ((const float4*)x)[(size_t)node * Dq + c4];
  float* s = sums + (size_t)gidx * D + (c4 << 2);
  atomicAdd(s + 0, v.x);
  atomicAdd(s + 1, v.y);
  atomicAdd(s + 2, v.z);
  atomicAdd(s + 3, v.w);
  if (c4 == 0) atomicAdd(&cnt[gidx], 1.f);
}

__global__ void pool_div_kernel(float* __restrict__ sums, const float* __restrict__ cnt,
                                int G_, int D) {
  const int i = blockIdx.x * blockDim.x + threadIdx.x;
  if (i >= G_ * D) return;
  sums[i] /= fmaxf(cnt[i / D], 1.f);
}

// ---- final head: out[g] = x[g,:] . W3 + b3, one wave per graph ----
__global__ void final_head_kernel(const float* __restrict__ x, const float* __restrict__ W3,
                                  const float* __restrict__ b3, float* __restrict__ out,
                                  int G_, int D) {
  const int g    = blockIdx.x * (blockDim.x >> 5) + (threadIdx.x >> 5);
  const int lane = threadIdx.x & 31;
  if (g >= G_) return;
  float s = 0.f;
  for (int j = lane; j < D; j += 32) s += x[(size_t)g * D + j] * W3[j];
#pragma unroll
  for (int off = 16; off; off >>= 1) s += __shfl_xor(s, off, 32);
  if (lane == 0) out[g] = s + b3[0];
}

extern "C" void kernel_launch(void* const* d_in, const int* in_sizes, int n_in,
                              void* d_out, int out_size, void* d_ws, size_t ws_size,
                              hipStream_t stream) {
  const float* x     = (const float*)d_in[0];
  const int*   ei    = (const int*)d_in[1];
  const int*   batch = (const int*)d_in[2];
  const float *Wl[3], *bl[3], *Wr[3], *br[3], *att[3], *bo[3], *gam[3], *bet[3];
  for (int L = 0; L < 3; ++L) {
    const int b = 3 + 8 * L;
    Wl[L]  = (const float*)d_in[b + 0];  bl[L]  = (const float*)d_in[b + 1];
    Wr[L]  = (const float*)d_in[b + 2];  br[L]  = (const float*)d_in[b + 3];
    att[L] = (const float*)d_in[b + 4];  bo[L]  = (const float*)d_in[b + 5];
    gam[L] = (const float*)d_in[b + 6];  bet[L] = (const float*)d_in[b + 7];
  }
  const float* W1 = (const float*)d_in[27]; const float* b1 = (const float*)d_in[28];
  const float* W2 = (const float*)d_in[29]; const float* b2 = (const float*)d_in[30];
  const float* W3 = (const float*)d_in[31]; const float* b3 = (const float*)d_in[32];
  float* out = (float*)d_out;

  const int EP = N_EDGES + N_NODES;   // edges + self loops
  float* ws = (float*)d_ws;
  size_t off = 0;
  auto walloc = [&](size_t nelem) { float* p = ws + off; off += nelem; return p; };
  float* xl     = walloc((size_t)N_NODES * 512);
  float* xr     = walloc((size_t)N_NODES * 512);
  float* agg    = walloc((size_t)N_NODES * 512);
  float* xcur   = walloc((size_t)N_NODES * 512);
  float* pbuf   = walloc((size_t)EP * 4);
  float* mmax   = walloc((size_t)N_NODES * 4);
  float* den    = walloc((size_t)N_NODES * 4);
  float* pooled = walloc((size_t)N_GRAPHS * HIDC);
  float* cnt    = walloc((size_t)N_GRAPHS);
  float* m1     = walloc((size_t)N_GRAPHS * HIDC);
  float* m2     = walloc((size_t)N_GRAPHS * (HIDC / 2));
  (void)ws_size; (void)n_in; (void)in_sizes; (void)out_size;

  const int layerH[3]   = {4, 4, 1};
  const int layerDin[3] = {DIM_IN, 512, 512};
  const float* lin = x;

  for (int L = 0; L < 3; ++L) {
    const int H = layerH[L], C = HIDC, D = H * C, din = layerDin[L];
    // node transforms: xl = lin@Wl+bl, xr = lin@Wr+br  (fp32 WMMA)
    dim3 gg(D / 64, N_NODES / 16);
    wmma_gemm_bias<<<gg, 128, 0, stream>>>(lin, Wl[L], bl[L], xl, N_NODES, din, D, 0);
    wmma_gemm_bias<<<gg, 128, 0, stream>>>(lin, Wr[L], br[L], xr, N_NODES, din, D, 0);
    // init segment buffers
    {
      size_t nm = (size_t)N_NODES * H;
      fill_f32<<<(unsigned)((nm + 255) / 256), 256, 0, stream>>>(mmax, -1e30f, nm);
      fill_f32<<<(unsigned)((nm + 255) / 256), 256, 0, stream>>>(den, 0.f, nm);
      size_t na = (size_t)N_NODES * D;
      fill_f32<<<(unsigned)((na + 255) / 256), 256, 0, stream>>>(agg, 0.f, na);
    }
    // edge phase
    edge_logits_kernel<<<(EP + 7) / 8, 256, 0, stream>>>(ei, xl, xr, att[L], pbuf, mmax,
                                                         N_EDGES, N_NODES, H, C);
    edge_softmax_kernel<<<(EP * H + 255) / 256, 256, 0, stream>>>(ei, mmax, pbuf, den,
                                                                  N_EDGES, N_NODES, H);
    edge_aggregate_kernel<<<(EP + 7) / 8, 256, 0, stream>>>(ei, xl, pbuf, den, agg,
                                                            N_EDGES, N_NODES, H, C);
    // bias + LayerNorm + ELU -> xcur
    node_ln_elu_kernel<<<(N_NODES + 7) / 8, 256, 0, stream>>>(agg, bo[L], gam[L], bet[L],
                                                              xcur, N_NODES, D);
    lin = xcur;
  }

  // global mean pool over graphs (xcur is [N,128] after layer 2)
  {
    size_t np = (size_t)N_GRAPHS * HIDC;
    fill_f32<<<(unsigned)((np + 255) / 256), 256, 0, stream>>>(pooled, 0.f, np);
    fill_f32<<<1, 64, 0, stream>>>(cnt, 0.f, (size_t)N_GRAPHS);
    pool_sum_kernel<<<(N_NODES * (HIDC / 4) + 255) / 256, 256, 0, stream>>>(
        xcur, batch, pooled, cnt, N_NODES, HIDC);
    pool_div_kernel<<<(N_GRAPHS * HIDC + 255) / 256, 256, 0, stream>>>(pooled, cnt,
                                                                       N_GRAPHS, HIDC);
  }

  // MLP head (fp32 WMMA GEMMs, fused ReLU)
  wmma_gemm_bias<<<dim3(HIDC / 64, N_GRAPHS / 16), 128, 0, stream>>>(
      pooled, W1, b1, m1, N_GRAPHS, HIDC, HIDC, 1);
  wmma_gemm_bias<<<dim3((HIDC / 2) / 64, N_GRAPHS / 16), 128, 0, stream>>>(
      m1, W2, b2, m2, N_GRAPHS, HIDC, HIDC / 2, 1);
  final_head_kernel<<<(N_GRAPHS + 7) / 8, 256, 0, stream>>>(m2, W3, b3, out,
                                                            N_GRAPHS, HIDC / 2);
}